// DistancePenalty_84731114815829
// MI455X (gfx1250) — compile-verified
//
#include <hip/hip_runtime.h>
#include <math.h>

typedef float v2f __attribute__((ext_vector_type(2)));
typedef float v8f __attribute__((ext_vector_type(8)));

#define N_ATOM   6144
#define N_FLAT   (3 * N_ATOM)
#define TILES    (N_ATOM / 16)      // 384

// workspace layout (float offsets)
#define WS_NORMS   0                // 6144 floats: squared norms per atom
#define WS_MOMSUM  6144             // 24 floats: per-block sum(x)
#define WS_MOMSQ   6168             // 24 floats: per-block sum(x^2)
#define WS_QVALS   6192             // 4 floats: order stats v[4607],v[4608],v[13823],v[13824]
#define WS_PEN     6208             // PEN_BLOCKS floats: per-block penalty partials
#define MOM_BLOCKS 24
#define PEN_BLOCKS 288
#define Q_BLOCKS   72
#define Q_CHUNK    4608             // 18 KB LDS chunk, 18432 = 4*4608

// ---------------------------------------------------------------------------
// K1: per-atom squared norms + moment partials (sum, sumsq) per block.
// 24 blocks x 256 threads == 6144 atoms exactly; each thread owns 3 floats,
// so moments cover all 18432 elements exactly once.
// ---------------------------------------------------------------------------
__global__ void norms_moments_kernel(const float* __restrict__ g,
                                     float* __restrict__ ws) {
    int i = blockIdx.x * blockDim.x + threadIdx.x;   // atom index
    float x = g[3 * i + 0];
    float y = g[3 * i + 1];
    float z = g[3 * i + 2];
    float nr = x * x + y * y + z * z;
    ws[WS_NORMS + i] = nr;

    __shared__ float ssum[256];
    __shared__ float ssq[256];
    ssum[threadIdx.x] = x + y + z;
    ssq[threadIdx.x]  = nr;                          // sum of squares == norm
    __syncthreads();
    for (int off = 128; off > 0; off >>= 1) {
        if ((int)threadIdx.x < off) {
            ssum[threadIdx.x] += ssum[threadIdx.x + off];
            ssq[threadIdx.x]  += ssq[threadIdx.x + off];
        }
        __syncthreads();
    }
    if (threadIdx.x == 0) {
        ws[WS_MOMSUM + blockIdx.x] = ssum[0];
        ws[WS_MOMSQ  + blockIdx.x] = ssq[0];
    }
}

// ---------------------------------------------------------------------------
// K2: pairwise penalty; the WMMA computes dist^2 tiles DIRECTLY:
//   A row m  = ( x_m,  y_m,  z_m,  1      )   (16x4, K3 slot = 1)
//   B col n  = (-2x_n, -2y_n, -2z_n, |x_n|^2)
//   C[m][n]  = |x_m|^2
//   D = A.B + C = |x_m|^2 - 2 x_m.x_n + |x_n|^2 = dist^2(m,n)
//
// Fragment layout (f32 16x16x4, wave32): lanes 0-15 carry K0(K=VGPR0)/K1,
// lanes 16-31 carry K2/K3. C/D: lane l -> N=l%16, VGPR r -> M=r+8*(l>=16).
// Tail per element: raw v_sqrt_f32, sub, max, cndmask-accumulate (branchless).
// ---------------------------------------------------------------------------
__global__ void pen_kernel(const float* __restrict__ g,
                           const float* __restrict__ norms,
                           float* __restrict__ partials) {
    const int lane       = threadIdx.x & 31;
    const int waveInBlk  = threadIdx.x >> 5;
    const int wavesPerBl = blockDim.x >> 5;
    const int gw         = blockIdx.x * wavesPerBl + waveInBlk;
    const int nw         = gridDim.x * wavesPerBl;
    const int half       = lane >> 4;     // 0: K0/K1 lanes, 1: K2/K3 lanes
    const int lm         = lane & 15;

    float acc = 0.0f;

    for (int p = gw; p < TILES * TILES; p += nw) {
        int ti = p / TILES;
        int tj = p - ti * TILES;
        if (tj < ti) continue;            // wave-uniform skip (EXEC stays full)

        int m = ti * 16 + lm;             // row atom for this lane
        int n = tj * 16 + lm;             // col atom for this lane

        float ax = g[3 * m + 0], ay = g[3 * m + 1], az = g[3 * m + 2];
        float bx = g[3 * n + 0], by = g[3 * n + 1], bz = g[3 * n + 2];
        float cn = norms[n];              // |x_n|^2 feeds the B fragment

        v2f a, b;
        a.x = half ? az : ax;             // K0 | K2
        a.y = half ? 1.0f : ay;           // K1 | K3(=1)
        b.x = half ? (-2.0f * bz) : (-2.0f * bx);
        b.y = half ? cn : (-2.0f * by);

        int rowbase = ti * 16 + half * 8;
        v8f c;
#pragma unroll
        for (int r = 0; r < 8; ++r)       // C = row norms (one load clause)
            c[r] = norms[rowbase + r];

        c = __builtin_amdgcn_wmma_f32_16x16x4_f32(
                /*neg_a=*/false, a, /*neg_b=*/false, b,
                /*c_mod=*/(short)0, c, /*reuse_a=*/false, /*reuse_b=*/false);

#pragma unroll
        for (int r = 0; r < 8; ++r) {
            int ig = rowbase + r;                 // global row index of c[r]
            float d = __builtin_amdgcn_sqrtf(fmaxf(c[r], 0.0f)); // raw v_sqrt_f32
            float term = fmaxf(0.8f - d, 0.0f);
            acc += (ig < n) ? term : 0.0f;        // strict upper triangle, cndmask
        }
    }

    // wave32 reduction
    for (int off = 16; off > 0; off >>= 1)
        acc += __shfl_down(acc, off, 32);

    __shared__ float swave[32];
    if (lane == 0) swave[waveInBlk] = acc;
    __syncthreads();
    if (threadIdx.x == 0) {
        float t = 0.0f;
        for (int w = 0; w < wavesPerBl; ++w) t += swave[w];  // fixed order
        partials[blockIdx.x] = t;
    }
}

// ---------------------------------------------------------------------------
// K3: exact order statistics via stable-rank counting.
// Thread i owns x_i, counts elements strictly below it (ties broken by
// index), streaming the array through LDS. Exactly one thread holds each
// rank, so the 4 target writes are race-free and deterministic.
// ---------------------------------------------------------------------------
__global__ void quantile_kernel(const float* __restrict__ g,
                                float* __restrict__ qvals) {
    const int i = blockIdx.x * blockDim.x + threadIdx.x;  // 0..18431
    const float v = g[i];
    int cnt = 0;

    __shared__ float chunk[Q_CHUNK];
    for (int c0 = 0; c0 < N_FLAT; c0 += Q_CHUNK) {
        for (int j = threadIdx.x; j < Q_CHUNK; j += blockDim.x)
            chunk[j] = g[c0 + j];
        __syncthreads();
#pragma unroll 8
        for (int j = 0; j < Q_CHUNK; ++j) {
            float u = chunk[j];
            int jg = c0 + j;
            cnt += (u < v) | ((u == v) & (jg < i));
        }
        __syncthreads();
    }

    if (cnt == 4607)  qvals[0] = v;
    if (cnt == 4608)  qvals[1] = v;
    if (cnt == 13823) qvals[2] = v;
    if (cnt == 13824) qvals[3] = v;
}

// ---------------------------------------------------------------------------
// K4: combine everything (single thread, fixed summation order).
// ---------------------------------------------------------------------------
__global__ void final_kernel(const float* __restrict__ ws,
                             float* __restrict__ out) {
    double pen = 0.0;
    for (int i = 0; i < PEN_BLOCKS; ++i) pen += (double)ws[WS_PEN + i];
    const double npairs = (double)N_ATOM * (double)(N_ATOM - 1) * 0.5;
    double punish_below = pen / npairs;

    double s = 0.0, sq = 0.0;
    for (int i = 0; i < MOM_BLOCKS; ++i) {
        s  += (double)ws[WS_MOMSUM + i];
        sq += (double)ws[WS_MOMSQ + i];
    }
    const double n = (double)N_FLAT;
    double var = (sq - s * s / n) / (n - 1.0);   // ddof=1 (torch.std)
    double sd  = sqrt(var);
    double std_punish = (sd - 1.75) * (sd - 1.75);

    // linear-interp quantiles: pos25 = 0.25*18431 = 4607.75,
    //                          pos75 = 0.75*18431 = 13823.25
    double v0 = ws[WS_QVALS + 0], v1 = ws[WS_QVALS + 1];
    double v2 = ws[WS_QVALS + 2], v3 = ws[WS_QVALS + 3];
    double q1 = v0 + 0.75 * (v1 - v0);
    double q3 = v2 + 0.25 * (v3 - v2);
    double d  = (q3 - q1) - 2.45;
    double iqr_punish = d * d;

    out[0] = (float)(punish_below + std_punish + iqr_punish);
}

extern "C" void kernel_launch(void* const* d_in, const int* in_sizes, int n_in,
                              void* d_out, int out_size, void* d_ws, size_t ws_size,
                              hipStream_t stream) {
    (void)in_sizes; (void)n_in; (void)out_size; (void)ws_size;
    const float* g  = (const float*)d_in[0];
    float*       ws = (float*)d_ws;
    float*       out = (float*)d_out;

    norms_moments_kernel<<<MOM_BLOCKS, 256, 0, stream>>>(g, ws);
    pen_kernel<<<PEN_BLOCKS, 256, 0, stream>>>(g, ws + WS_NORMS, ws + WS_PEN);
    quantile_kernel<<<Q_BLOCKS, 256, 0, stream>>>(g, ws + WS_QVALS);
    final_kernel<<<1, 1, 0, stream>>>(ws, out);
}